// GaussianSampler_19920058319547
// MI455X (gfx1250) — compile-verified
//
#include <hip/hip_runtime.h>
#include <hip/hip_bf16.h>
#include <math.h>

#define NG   1024
#define IMGW 256
#define IMGH 256
#define NPIX (IMGW * IMGH)

typedef float v2f __attribute__((ext_vector_type(2)));
typedef float v8f __attribute__((ext_vector_type(8)));

// ---------------------------------------------------------------------------
// Kernel 1: per-gaussian preprocess + depth bitonic sort (single 1024-thread
// block). Writes radii (original order) to d_out tail, sorted SoA to ws:
//   ws[0*NG..] px, [1] py, [2] cA, [3] cB, [4] cC, [5] op*valid,
//   [6] r, [7] g, [8] b     (9*1024 floats = 36KB)
// ---------------------------------------------------------------------------
__global__ __launch_bounds__(1024)
void gs_preprocess_sort(const float* __restrict__ means3D,
                        const float* __restrict__ colors,
                        const float* __restrict__ opac,
                        const float* __restrict__ scales,
                        const float* __restrict__ rots,
                        const float* __restrict__ vm,
                        const float* __restrict__ pm,
                        float* __restrict__ ws,
                        int* __restrict__ radii_out)
{
    __shared__ float s_px[NG], s_py[NG], s_cA[NG], s_cB[NG], s_cC[NG];
    __shared__ float s_op[NG], s_r[NG], s_g[NG], s_b[NG];
    __shared__ float s_key[NG];
    __shared__ int   s_idx[NG];

    const int i = threadIdx.x;

    // ---- quaternion -> rotation ----
    float qr = rots[i * 4 + 0], qx = rots[i * 4 + 1];
    float qy = rots[i * 4 + 2], qz = rots[i * 4 + 3];
    float qn = rsqrtf(qr * qr + qx * qx + qy * qy + qz * qz);
    qr *= qn; qx *= qn; qy *= qn; qz *= qn;
    float R00 = 1.f - 2.f * (qy * qy + qz * qz), R01 = 2.f * (qx * qy - qr * qz), R02 = 2.f * (qx * qz + qr * qy);
    float R10 = 2.f * (qx * qy + qr * qz), R11 = 1.f - 2.f * (qx * qx + qz * qz), R12 = 2.f * (qy * qz - qr * qx);
    float R20 = 2.f * (qx * qz - qr * qy), R21 = 2.f * (qy * qz + qr * qx), R22 = 1.f - 2.f * (qx * qx + qy * qy);

    // ---- M = R * scales (column scaling), cov3d = M M^T ----
    float s0 = scales[i * 3 + 0], s1 = scales[i * 3 + 1], s2 = scales[i * 3 + 2];
    float M00 = R00 * s0, M01 = R01 * s1, M02 = R02 * s2;
    float M10 = R10 * s0, M11 = R11 * s1, M12 = R12 * s2;
    float M20 = R20 * s0, M21 = R21 * s1, M22 = R22 * s2;
    float c00 = M00 * M00 + M01 * M01 + M02 * M02;
    float c01 = M00 * M10 + M01 * M11 + M02 * M12;
    float c02 = M00 * M20 + M01 * M21 + M02 * M22;
    float c11 = M10 * M10 + M11 * M11 + M12 * M12;
    float c12 = M10 * M20 + M11 * M21 + M12 * M22;
    float c22 = M20 * M20 + M21 * M21 + M22 * M22;

    // ---- camera transform ----
    float m0 = means3D[i * 3 + 0], m1 = means3D[i * 3 + 1], m2 = means3D[i * 3 + 2];
    float t0 = m0 * vm[0]  + m1 * vm[4]  + m2 * vm[8]  + vm[12];
    float t1 = m0 * vm[1]  + m1 * vm[5]  + m2 * vm[9]  + vm[13];
    float tz = m0 * vm[2]  + m1 * vm[6]  + m2 * vm[10] + vm[14];

    // ---- projection ----
    float ph0 = m0 * pm[0] + m1 * pm[4] + m2 * pm[8]  + pm[12];
    float ph1 = m0 * pm[1] + m1 * pm[5] + m2 * pm[9]  + pm[13];
    float ph3 = m0 * pm[3] + m1 * pm[7] + m2 * pm[11] + pm[15];
    float invw = 1.f / (ph3 + 1e-7f);
    float px = ((ph0 * invw + 1.f) * (float)IMGW - 1.f) * 0.5f;
    float py = ((ph1 * invw + 1.f) * (float)IMGH - 1.f) * 0.5f;

    // ---- EWA splat: J, T_, cov2d ----
    const float fx = 256.f, fy = 256.f, lim = 0.65f;
    float itz  = 1.f / tz;
    float txz  = fminf(fmaxf(t0 * itz, -lim), lim) * tz;
    float tyz  = fminf(fmaxf(t1 * itz, -lim), lim) * tz;
    float J00  = fx * itz;
    float J02  = -fx * txz * itz * itz;
    float J11  = fy * itz;
    float J12  = -fy * tyz * itz * itz;
    // T_[i][k] = sum_j J[i][j] * Vrot[k][j],  Vrot[k][j] = vm[k*4+j]
    float T00 = J00 * vm[0] + J02 * vm[2];
    float T01 = J00 * vm[4] + J02 * vm[6];
    float T02 = J00 * vm[8] + J02 * vm[10];
    float T10 = J11 * vm[1] + J12 * vm[2];
    float T11 = J11 * vm[5] + J12 * vm[6];
    float T12 = J11 * vm[9] + J12 * vm[10];
    // tmp[i][k] = sum_j T[i][j] * cov3d[j][k]
    float u0 = T00 * c00 + T01 * c01 + T02 * c02;
    float u1 = T00 * c01 + T01 * c11 + T02 * c12;
    float u2 = T00 * c02 + T01 * c12 + T02 * c22;
    float w0 = T10 * c00 + T11 * c01 + T12 * c02;
    float w1 = T10 * c01 + T11 * c11 + T12 * c12;
    float w2 = T10 * c02 + T11 * c12 + T12 * c22;
    float cov00 = u0 * T00 + u1 * T01 + u2 * T02;
    float cov01 = u0 * T10 + u1 * T11 + u2 * T12;
    float cov11 = w0 * T10 + w1 * T11 + w2 * T12;

    float a = cov00 + 0.3f;
    float c = cov11 + 0.3f;
    float b = cov01;
    float det = a * c - b * b;
    bool valid = (tz > 0.2f) && (det > 0.f);
    float det_s = (det == 0.f) ? 1.f : det;
    float conA = c / det_s, conB = -b / det_s, conC = a / det_s;

    float mid  = 0.5f * (a + c);
    float lam1 = mid + sqrtf(fmaxf(0.1f, mid * mid - det));
    radii_out[i] = valid ? (int)ceilf(3.f * sqrtf(lam1)) : 0;

    // zero everything for invalid -> power==0, alpha==0 downstream, no NaNs
    s_px[i] = valid ? px : 0.f;
    s_py[i] = valid ? py : 0.f;
    s_cA[i] = valid ? conA : 0.f;
    s_cB[i] = valid ? conB : 0.f;
    s_cC[i] = valid ? conC : 0.f;
    s_op[i] = valid ? opac[i] : 0.f;
    s_r[i]  = colors[i * 3 + 0];
    s_g[i]  = colors[i * 3 + 1];
    s_b[i]  = colors[i * 3 + 2];
    s_key[i] = valid ? tz : __builtin_inff();
    s_idx[i] = i;
    __syncthreads();

    // ---- bitonic sort (key ascending) ----
    for (int k = 2; k <= NG; k <<= 1) {
        for (int j = k >> 1; j > 0; j >>= 1) {
            int ixj = i ^ j;
            if (ixj > i) {
                float k0 = s_key[i], k1 = s_key[ixj];
                bool up = ((i & k) == 0);
                if ((k0 > k1) == up) {
                    s_key[i] = k1; s_key[ixj] = k0;
                    int t = s_idx[i]; s_idx[i] = s_idx[ixj]; s_idx[ixj] = t;
                }
            }
            __syncthreads();
        }
    }

    int s = s_idx[i];
    ws[0 * NG + i] = s_px[s];
    ws[1 * NG + i] = s_py[s];
    ws[2 * NG + i] = s_cA[s];
    ws[3 * NG + i] = s_cB[s];
    ws[4 * NG + i] = s_cC[s];
    ws[5 * NG + i] = s_op[s];
    ws[6 * NG + i] = s_r[s];
    ws[7 * NG + i] = s_g[s];
    ws[8 * NG + i] = s_b[s];
}

// ---------------------------------------------------------------------------
// Kernel 2: tile raster. One block per 16x16 pixel tile, 512 threads = 16
// waves; wave w owns pixel row v=w, lane&15 = pixel column. Power is computed
// as a bilinear form via V_WMMA_F32_16X16X4_F32:
//   A (M=16 gaussians x K)  : gaussian features from LDS
//   B (K x N=16 pixels)     : pixel features [1,u,u^2,v,v^2,uv,0,0] (K=8 => 2 wmma)
//   D VGPR j, lane L        : gaussian gb+j+8*(L>>4), pixel L&15
// Gaussian loop unrolled x2 (4 WMMAs in flight over each serial blend).
// ---------------------------------------------------------------------------

// Serial front-to-back blend of one 16-gaussian block, then cross-half merge.
__device__ __forceinline__
void blend16(const v8f d, int gb, int h, const float4* __restrict__ cop,
             float& Tacc, float& Cr, float& Cg, float& Cb)
{
    float Tl = 1.f, lr = 0.f, lg = 0.f, lb = 0.f;
    const int gg = gb + 8 * h;
    #pragma unroll
    for (int j = 0; j < 8; ++j) {
        float4 cg = cop[gg + j];                 // one ds_load_b128
        float pw = d[j];
        float al = fminf(0.99f, cg.w * __expf(pw));
        al = (pw > 0.f || al < 0.0039215686f) ? 0.f : al;
        float wj = Tl * al;
        lr += wj * cg.x;
        lg += wj * cg.y;
        lb += wj * cg.z;
        Tl *= 1.f - al;
    }
    // merge low half (gaussians gb..gb+7) with high half (gb+8..gb+15)
    float oT = __shfl_xor(Tl, 16, 32);
    float oR = __shfl_xor(lr, 16, 32);
    float oG = __shfl_xor(lg, 16, 32);
    float oB = __shfl_xor(lb, 16, 32);
    float lT = h ? oT : Tl, hT = h ? Tl : oT;
    float lR = h ? oR : lr, hR = h ? lr : oR;
    float lG = h ? oG : lg, hG = h ? lg : oG;
    float lB = h ? oB : lb, hB = h ? lb : oB;
    Cr += Tacc * (lR + lT * hR);
    Cg += Tacc * (lG + lT * hG);
    Cb += Tacc * (lB + lT * hB);
    Tacc *= lT * hT;
}

__global__ __launch_bounds__(512)
void gs_raster(const float* __restrict__ ws,
               const float* __restrict__ bg,
               float* __restrict__ out)
{
    __shared__ float  feat[NG][8];  // 32KB: per-tile gaussian power features
    __shared__ float4 cop[NG];      // 16KB: {r,g,b,op} packed for 1x b128/blend step

    const int tid = threadIdx.x;
    const int tx = blockIdx.x & 15, ty = blockIdx.x >> 4;
    const float cx = (float)(tx * 16) + 7.5f;
    const float cy = (float)(ty * 16) + 7.5f;

    for (int g = tid; g < NG; g += 512) {
        __builtin_prefetch(&ws[0 * NG + g + 512], 0, 3);  // global_prefetch_b8
        float px = ws[0 * NG + g], py = ws[1 * NG + g];
        float cA = ws[2 * NG + g], cB = ws[3 * NG + g], cC = ws[4 * NG + g];
        float gx = px - cx, gy = py - cy;
        float hA = -0.5f * cA, hC = -0.5f * cC, nB = -cB;
        feat[g][0] = hA * gx * gx + hC * gy * gy + nB * gx * gy; // * 1
        feat[g][1] = cA * gx + cB * gy;                          // * u
        feat[g][2] = hA;                                         // * u^2
        feat[g][3] = cC * gy + cB * gx;                          // * v
        feat[g][4] = hC;                                         // * v^2
        feat[g][5] = nB;                                         // * u*v
        feat[g][6] = 0.f;
        feat[g][7] = 0.f;
        float4 cg;
        cg.x = ws[6 * NG + g];
        cg.y = ws[7 * NG + g];
        cg.z = ws[8 * NG + g];
        cg.w = ws[5 * NG + g];
        cop[g] = cg;
    }
    __syncthreads();

    const int w    = tid >> 5;        // wave id == tile-local pixel row
    const int lane = tid & 31;
    const int h    = lane >> 4;       // K-half / gaussian-half selector
    const int m    = lane & 15;       // pixel column (N) and A-row (M)
    const float uu = (float)m - 7.5f;
    const float vv = (float)w - 7.5f;

    // B matrix (K x 16 pixels): lanes0-15 hold rows K0/K1, lanes16-31 K2/K3
    v2f B0, B1;
    B0.x = h ? uu * uu   : 1.f;       // K2 : K0
    B0.y = h ? vv        : uu;        // K3 : K1
    B1.x = h ? 0.f       : vv * vv;   // K6 : K4
    B1.y = h ? 0.f       : uu * vv;   // K7 : K5

    float Tacc = 1.f, Cr = 0.f, Cg = 0.f, Cb = 0.f;

    for (int gb = 0; gb < NG; gb += 32) {
        // A fragments for two consecutive 16-gaussian blocks
        v2f A0a = *(const v2f*)&feat[gb + m][2 * h];
        v2f A1a = *(const v2f*)&feat[gb + m][4 + 2 * h];
        v2f A0b = *(const v2f*)&feat[gb + 16 + m][2 * h];
        v2f A1b = *(const v2f*)&feat[gb + 16 + m][4 + 2 * h];

        v8f d0 = {};
        d0 = __builtin_amdgcn_wmma_f32_16x16x4_f32(false, A0a, false, B0,
                                                   (short)0, d0, false, false);
        d0 = __builtin_amdgcn_wmma_f32_16x16x4_f32(false, A1a, false, B1,
                                                   (short)0, d0, false, false);
        v8f d1 = {};
        d1 = __builtin_amdgcn_wmma_f32_16x16x4_f32(false, A0b, false, B0,
                                                   (short)0, d1, false, false);
        d1 = __builtin_amdgcn_wmma_f32_16x16x4_f32(false, A1b, false, B1,
                                                   (short)0, d1, false, false);

        blend16(d0, gb,      h, cop, Tacc, Cr, Cg, Cb);
        blend16(d1, gb + 16, h, cop, Tacc, Cr, Cg, Cb);
    }

    if (h == 0) {
        int x = tx * 16 + m, y = ty * 16 + w;
        int p = y * IMGW + x;
        __builtin_nontemporal_store(Cr + Tacc * bg[0], &out[0 * NPIX + p]);
        __builtin_nontemporal_store(Cg + Tacc * bg[1], &out[1 * NPIX + p]);
        __builtin_nontemporal_store(Cb + Tacc * bg[2], &out[2 * NPIX + p]);
    }
}

// ---------------------------------------------------------------------------
extern "C" void kernel_launch(void* const* d_in, const int* in_sizes, int n_in,
                              void* d_out, int out_size, void* d_ws, size_t ws_size,
                              hipStream_t stream)
{
    const float* means3D = (const float*)d_in[0];
    // d_in[1] = means2D (unused by reference math)
    const float* colors  = (const float*)d_in[2];
    const float* opac    = (const float*)d_in[3];
    const float* scales  = (const float*)d_in[4];
    const float* rots    = (const float*)d_in[5];
    const float* vm      = (const float*)d_in[6];
    const float* pm      = (const float*)d_in[7];
    const float* bg      = (const float*)d_in[8];

    float* out   = (float*)d_out;
    int*   radii = (int*)d_out + 3 * NPIX;   // radii stored after color, as i32 bits
    float* ws    = (float*)d_ws;             // 9*1024 floats

    gs_preprocess_sort<<<1, 1024, 0, stream>>>(means3D, colors, opac, scales,
                                               rots, vm, pm, ws, radii);
    gs_raster<<<256, 512, 0, stream>>>(ws, bg, out);
}